// DepthRouter_601295421732
// MI455X (gfx1250) — compile-verified
//
#include <hip/hip_runtime.h>
#include <hip/hip_bf16.h>

// DepthRouter for MI455X (gfx1250).
// Roofline: x is 128MB read once -> ~5.5us floor at 23.3 TB/s; 67 MFLOP of
// dot products is free. Pipeline: GLOBAL_LOAD_ASYNC_TO_LDS_B128 (coalesced
// 512B row segments, ASYNCcnt-tracked, double-buffered) -> ds_load_b64 in
// WMMA layout -> V_WMMA_F32_16X16X4_F32 (exact fp32 reduction).

typedef float v2f __attribute__((ext_vector_type(2)));
typedef float v8f __attribute__((ext_vector_type(8)));

#define B_DIM 4
#define S_DIM 4096
#define D_DIM 2048
#define ROWS  (B_DIM * S_DIM)   // 16384
#define K_SEL (S_DIM / 2)       // 2048 (CAPACITY = 0.5)

#define WAVES_PER_BLK 4
#define CHUNK_K  128                       // K-chunk per double-buffer stage
#define NCHUNK   (D_DIM / CHUNK_K)         // 16
#define ROW_PAD  132                       // 128 floats + 4 pad -> 528B row
                                           // stride = bank offset 4 per row:
                                           // 16 rows hit 16 distinct banks
#define BUF_FLTS (16 * ROW_PAD)            // 2112 floats = 8448 B per buffer

__device__ __forceinline__ void async_load_row(const float* g, unsigned lds_off) {
    // GLOBAL_LOAD_ASYNC_TO_LDS_B128: each lane copies 16B from global address g to LDS offset lds_off; whole
    // wave moves one contiguous 512B row segment. Tracked on ASYNCcnt.
    asm volatile("global_load_async_to_lds_b128 %0, %1, off"
                 :: "v"(lds_off), "v"(g) : "memory");
}

// ---------------------------------------------------------------------------
// Kernel 1: logits + weights. One wave computes a 16-row tile, K = 2048,
// via 512 WMMA f32 16x16x4 accumulations fed from a double-buffered async
// global->LDS pipeline.
//   A (16x4) = gate_w chunk replicated over M:  lane L: (w[k0+2h], w[k0+1+2h])
//   B (4x16) = 16 x-rows as columns:  lane L: (x[row L&15][k0+2h], ...+1)
//   D rows identical = logits; D[0][n] read from c[0], lanes 0..15.
// out: out[row] = logit (temp; kernel 2 overwrites with mask),
//      out[ROWS+row] = sigmoid(logit).
// ---------------------------------------------------------------------------
__global__ __launch_bounds__(32 * WAVES_PER_BLK)
void gate_logits_wmma(const float* __restrict__ x,
                      const float* __restrict__ gw,
                      float* __restrict__ out)
{
    __shared__ v2f   WB2[D_DIM / 2];                    // 8KB gate_w
    __shared__ float XBUF[WAVES_PER_BLK][2][BUF_FLTS];  // 4 x 2 x 8448B

    const int tid = threadIdx.x;
    for (int j = tid; j < D_DIM / 2; j += 32 * WAVES_PER_BLK)
        WB2[j] = *(const v2f*)(gw + 2 * j);
    __syncthreads();

    const int lane     = tid & 31;
    const int wave     = tid >> 5;
    const int tile     = blockIdx.x * WAVES_PER_BLK + wave;  // 16 rows/tile
    const int row_base = tile * 16;
    const int n        = lane & 15;   // column (x-row) index in B
    const int h        = lane >> 4;   // half-wave K split

    // Per-buffer LDS byte offsets (low 32 bits of flat addr == LDS offset),
    // plus this lane's 16B slot within a row segment.
    const unsigned lbuf0 = (unsigned)(uintptr_t)&XBUF[wave][0][0] + lane * 16u;
    const unsigned lbuf1 = (unsigned)(uintptr_t)&XBUF[wave][1][0] + lane * 16u;

    auto issue_chunk = [&](int ch, int sel) {
        const float* g = x + (size_t)row_base * D_DIM + ch * CHUNK_K + lane * 4;
        unsigned l = sel ? lbuf1 : lbuf0;
        #pragma unroll
        for (int r = 0; r < 16; ++r) {       // one coalesced 512B row each
            async_load_row(g, l);
            g += D_DIM;                      // next x row (8KB stride)
            l += ROW_PAD * 4;                // next padded LDS row (528B)
        }
    };

    v8f c = {};                              // f32 accumulator tile

    issue_chunk(0, 0);
    for (int ch = 0; ch < NCHUNK; ++ch) {
        const int sel = ch & 1;
        if (ch + 1 < NCHUNK) {
            // Buffer (ch+1)&1 was last read in the previous iteration; make sure all
            // ds reads retired before the async DMA overwrites it (ds and
            // async LDS paths are mutually unordered).
            asm volatile("s_wait_dscnt 0x0" ::: "memory");
            issue_chunk(ch + 1, sel ^ 1);
            // 32 async ops outstanding; <=16 left means chunk `ch` resident.
            asm volatile("s_wait_asynccnt 0x10" ::: "memory");
        } else {
            asm volatile("s_wait_asynccnt 0x0" ::: "memory");
        }

        const float* bp = &XBUF[wave][sel][n * ROW_PAD + 2 * h];
        #pragma unroll
        for (int kk = 0; kk < CHUNK_K / 4; ++kk) {
            v2f a = WB2[(ch * (CHUNK_K / 2)) + 2 * kk + h];  // ds broadcast
            v2f b = *(const v2f*)(bp + 4 * kk);              // ds_load_b64
            c = __builtin_amdgcn_wmma_f32_16x16x4_f32(
                    false, a, false, b, (short)0, c, false, false);
        }
    }

    if (lane < 16) {
        float logit = c[0];                  // D[0][n]
        int row = row_base + n;
        out[row]        = logit;                              // temp logit
        out[ROWS + row] = 1.0f / (1.0f + __expf(-logit));     // weight
    }
}

// ---------------------------------------------------------------------------
// Kernel 2: exact top-k (k = 2048) per batch row. Radix bit-search on the
// order-preserving uint key finds the exact k-th largest logit; ties are
// selected smallest-index-first (matches jax.lax.top_k stability).
// Reads logits from out[b*S..], writes mask in place. Deterministic.
// ---------------------------------------------------------------------------
__global__ __launch_bounds__(256)
void topk_mask(float* __restrict__ out)
{
    __shared__ unsigned int keys[S_DIM];   // 16KB
    __shared__ unsigned int red[256];
    const int tid = threadIdx.x;
    float* logits = out + blockIdx.x * S_DIM;
    const int CPT = S_DIM / 256;           // 16 per thread

    for (int j = 0; j < CPT; ++j) {
        int i = j * 256 + tid;
        unsigned int u = __float_as_uint(logits[i]);
        keys[i] = (u & 0x80000000u) ? ~u : (u | 0x80000000u);
    }
    __syncthreads();

    unsigned int prefix = 0u;
    for (int bit = 31; bit >= 0; --bit) {
        unsigned int cand = prefix | (1u << bit);
        unsigned int cnt = 0;
        for (int j = 0; j < CPT; ++j)
            cnt += (keys[j * 256 + tid] >= cand) ? 1u : 0u;
        red[tid] = cnt;
        __syncthreads();
        for (int off = 128; off > 0; off >>= 1) {
            if (tid < off) red[tid] += red[tid + off];
            __syncthreads();
        }
        unsigned int total = red[0];
        __syncthreads();
        if (total >= K_SEL) prefix = cand;   // uniform update
    }

    unsigned int cnt = 0;
    for (int j = 0; j < CPT; ++j)
        cnt += (keys[j * 256 + tid] > prefix) ? 1u : 0u;
    red[tid] = cnt;
    __syncthreads();
    for (int off = 128; off > 0; off >>= 1) {
        if (tid < off) red[tid] += red[tid + off];
        __syncthreads();
    }
    unsigned int G = red[0];
    __syncthreads();
    unsigned int E = K_SEL - G;              // #ties accepted, index order

    unsigned int eq = 0;
    for (int i = tid * CPT; i < (tid + 1) * CPT; ++i)
        eq += (keys[i] == prefix) ? 1u : 0u;
    red[tid] = eq;
    __syncthreads();
    for (int off = 1; off < 256; off <<= 1) {
        unsigned int v = (tid >= off) ? red[tid - off] : 0u;
        __syncthreads();
        red[tid] += v;
        __syncthreads();
    }
    unsigned int run = red[tid] - eq;        // exclusive prefix

    for (int i = tid * CPT; i < (tid + 1) * CPT; ++i) {
        unsigned int kv = keys[i];
        float m;
        if (kv > prefix)       m = 1.0f;
        else if (kv == prefix) { m = (run < E) ? 1.0f : 0.0f; ++run; }
        else                   m = 0.0f;
        logits[i] = m;                       // overwrite logit with mask
    }
}

// ---------------------------------------------------------------------------
// Kernel 3: unbiased variance of all 16384 weights, fixed-order f64 tree
// reduction (deterministic). Writes out[2*ROWS].
// ---------------------------------------------------------------------------
__global__ __launch_bounds__(256)
void weight_var(float* __restrict__ out)
{
    __shared__ double s1[256];
    __shared__ double s2[256];
    const int tid = threadIdx.x;
    const float* w = out + ROWS;
    double a = 0.0, b = 0.0;
    for (int j = 0; j < ROWS / 256; ++j) {
        double v = (double)w[j * 256 + tid];
        a += v; b += v * v;
    }
    s1[tid] = a; s2[tid] = b;
    __syncthreads();
    for (int off = 128; off > 0; off >>= 1) {
        if (tid < off) { s1[tid] += s1[tid + off]; s2[tid] += s2[tid + off]; }
        __syncthreads();
    }
    if (tid == 0) {
        double nn = (double)ROWS;
        out[2 * ROWS] = (float)((s2[0] - s1[0] * s1[0] / nn) / (nn - 1.0));
    }
}

extern "C" void kernel_launch(void* const* d_in, const int* in_sizes, int n_in,
                              void* d_out, int out_size, void* d_ws, size_t ws_size,
                              hipStream_t stream)
{
    const float* x  = (const float*)d_in[0];   // [4,4096,2048] f32
    const float* gw = (const float*)d_in[1];   // [2048] f32
    float* out = (float*)d_out;                // mask | weights | aux

    // 1024 tiles of 16 rows; 4 waves/block -> 256 blocks for WGP coverage.
    gate_logits_wmma<<<ROWS / 16 / WAVES_PER_BLK, 32 * WAVES_PER_BLK, 0, stream>>>(x, gw, out);
    topk_mask<<<B_DIM, 256, 0, stream>>>(out);
    weight_var<<<1, 256, 0, stream>>>(out);
    (void)in_sizes; (void)n_in; (void)out_size; (void)d_ws; (void)ws_size;
}